// CompProbModel_42691974922925
// MI455X (gfx1250) — compile-verified
//
#include <hip/hip_runtime.h>
#include <math.h>

typedef __attribute__((ext_vector_type(2))) float v2f;
typedef __attribute__((ext_vector_type(8))) float v8f;

#define NXc 120
#define NYc 55
#define NTc 40
#define NPc 10
#define NNc 16   // padded player/column dimension for WMMA
#define Gc  10.72468f

__device__ __forceinline__ float field_x(int ix) { return 0.5f + (float)ix; }
__device__ __forceinline__ float field_y(int iy) { return (iy == 0) ? -0.2f : (-0.5f + (float)iy); }
__device__ __forceinline__ float tgrid(int s)    { return 0.1f + 0.1f * (float)s; }

__global__ __launch_bounds__(64) void comp_prob_kernel(
    const float* __restrict__ frame,
    const float* __restrict__ s_max_p,
    const float* __restrict__ a_max_p,
    const float* __restrict__ sigma_p,
    const float* __restrict__ lam_off_p,
    const float* __restrict__ lam_def_p,
    float* __restrict__ out)
{
    const int b   = blockIdx.x;
    const int tid = threadIdx.x;
    const float* fr = frame + b * NPc * 13;

    const float s_max  = s_max_p[0];
    const float a_max  = a_max_p[0];
    const float sigma  = sigma_p[0];
    const float lamO   = lam_off_p[0];
    const float lamD   = lam_def_p[0];
    const float reax   = s_max / a_max;
    const float sigc   = 3.14f / (1.732f * sigma);

    __shared__ float sh_rlx[NPc], sh_rly[NPc], sh_rvx[NPc], sh_rvy[NPc], sh_lam[NPc];
    __shared__ float sh_pnorm[NTc][NPc];
    __shared__ float sh_allt[NTc];
    __shared__ float sh_shift[NTc];
    __shared__ float sh_ind[NTc][NNc];   // zero-padded ind_dt[s][n], n in [0,16)

    // scalars derived from frame row 0 (uniform across the block)
    int tof = (int)rintf(fr[12]) - 1;              // jnp.round (RNE) then -1
    tof = min(max(tof, 0), NTc - 1);               // JAX gather clamp
    int bex = (int)fr[10];                         // astype(int32) truncation
    int bey = (int)fr[11];
    int b_idx = (bey + 1) * NXc + bex;
    b_idx = min(max(b_idx, 0), NXc * NYc - 1);     // JAX gather clamp
    const float bsx = fr[8], bsy = fr[9];

    // Phase 0: per-player reacted kinematics
    if (tid < NPc) {
        const float* r = fr + tid * 13;
        float x = r[1], y = r[2], vx = r[3], vy = r[4], ax = r[5], ay = r[6], team = r[7];
        float xr = x + vx * reax + 0.5f * ax * reax * reax;
        float yr = y + vy * reax + 0.5f * ay * reax * reax;
        sh_rlx[tid] = (float)(int)xr;              // astype(int32).astype(f32)
        sh_rly[tid] = (float)(int)yr;
        sh_rvx[tid] = ax * reax + vx;
        sh_rvy[tid] = ay * reax + vy;
        sh_lam[tid] = lamO * team + lamD * (1.0f - team);
    }
    __syncthreads();

    // Phase 1: one thread per substep s along the ball trajectory (f=b_idx, t=tof)
    if (tid < NTc) {
        const int s   = tid;
        const float Ts = tgrid(s);
        const float Tt = tgrid(tof);

        float fx = field_x(b_idx % NXc);
        float fy = field_y(b_idx / NXc);
        float vxb = (fx - bsx) / Tt;
        float vyb = (fy - bsy) / Tt;
        int tx = (int)rintf(fminf(fmaxf(bsx + vxb * Ts, 0.0f), (float)(NXc - 1)));
        int ty = (int)rintf(fminf(fmaxf(bsy + vyb * Ts, 0.0f), (float)(NYc - 1)));
        float px = field_x(tx);
        float py = field_y(ty);

        float vz0  = Tt * Gc * 0.5f;
        float z    = 2.0f + vz0 * Ts - 0.5f * Gc * Ts * Ts;
        float lamz = (z < 3.0f && z > 0.0f) ? 1.0f : 0.0f;

        float pt[NPc];
        float ssum = 0.0f;
        for (int p = 0; p < NPc; ++p) {
            float dx = px - sh_rlx[p];
            float dy = py - sh_rly[p];
            float dmag = sqrtf(dx * dx + dy * dy);
            float s0 = (dx * sh_rvx[p] + dy * sh_rvy[p]) / dmag;
            s0 = fminf(fmaxf(s0, -s_max), s_max);
            float t_lt = (s_max - s0) / a_max;
            float d_lt = t_lt * (s0 + s_max) * 0.5f;
            if (d_lt > dmag)
                t_lt = -s0 / a_max + sqrtf((s0 / a_max) * (s0 / a_max) + 2.0f * dmag / a_max);
            d_lt = fmaxf(fminf(d_lt, dmag), 0.0f);
            float t_tot = reax + t_lt + (dmag - d_lt) / s_max;
            float arg   = sigc * (Ts - t_tot);
            float pint  = 1.0f / (1.0f + expf(-arg));
            float v = pint * lamz;
            pt[p] = v;
            ssum += v;
        }
        float inv  = 1.0f / fmaxf(1.0f, ssum);
        float allt = 0.0f;
        for (int p = 0; p < NPc; ++p) {
            float pn = pt[p] * inv;
            sh_pnorm[s][p] = pn;
            allt += pn;
        }
        sh_allt[s] = allt;
    }
    __syncthreads();

    // Phase 2: cumprod over s, then roll-by-1 (reference semantics, incl. t==0 override)
    if (tid == 0) {
        float rem[NTc];
        float r = 1.0f;
        for (int s = 0; s < NTc; ++s) { r *= (1.0f - sh_allt[s]); rem[s] = r; }
        for (int s = 0; s < NTc; ++s)
            sh_shift[s] = (tof == 0) ? 1.0f : rem[(s + NTc - 1) % NTc];
    }
    __syncthreads();

    // Phase 2.5: materialize zero-padded ind_dt[s][n] so the WMMA loop is
    // branch-free (no lane-divergent EXEC save/restore around DS loads).
    if (tid < NTc) {
        const int s = tid;
        const float sh = sh_shift[s];
        #pragma unroll
        for (int n = 0; n < NNc; ++n)
            sh_ind[s][n] = (n < NPc) ? sh * sh_pnorm[s][n] * sh_lam[n] : 0.0f;
    }
    __syncthreads();

    // Phase 3: triangular contraction einsum('ts,sp->tp') as V_WMMA_F32_16X16X4_F32.
    // Wave 0 only (uniform branch -> EXEC all ones within the wave).
    if (tid < 32) {
        const int lane = tid;
        const int half = lane >> 4;      // K-pair selector for A/B fragments
        const int m    = lane & 15;      // A row (t - t0)
        const int n    = lane & 15;      // B/D column (player)
        const int t0   = tof & ~15;      // 16-row t-tile containing tof
        const int tm   = t0 + m;

        v8f c = {};
        #pragma unroll
        for (int k = 0; k < NTc / 4; ++k) {
            const int s0i = 4 * k + 2 * half;
            const int s1i = s0i + 1;
            v2f a, bf;
            a.x = (s0i <= tm) ? 1.0f : 0.0f;   // tril[t, s]
            a.y = (s1i <= tm) ? 1.0f : 0.0f;
            bf.x = sh_ind[s0i][n];
            bf.y = sh_ind[s1i][n];
            // 8 args: (neg_a, A, neg_b, B, c_mod, C, reuse_a, reuse_b)
            c = __builtin_amdgcn_wmma_f32_16x16x4_f32(
                    false, a, false, bf, (short)0, c, false, false);
        }
        // D layout: lanes 0-15 hold rows t0+0..7 in c[0..7]; lanes 16-31 rows t0+8..15
        if (n < NPc) {
            #pragma unroll
            for (int j = 0; j < 8; ++j) {
                int row = t0 + half * 8 + j;
                if (row == tof) out[b * NPc + n] = c[j];
            }
        }
    }
}

extern "C" void kernel_launch(void* const* d_in, const int* in_sizes, int n_in,
                              void* d_out, int out_size, void* d_ws, size_t ws_size,
                              hipStream_t stream) {
    const float* frame = (const float*)d_in[0];
    int B = in_sizes[0] / (NPc * 13);
    if (B < 1) B = 1;
    comp_prob_kernel<<<dim3(B), dim3(64), 0, stream>>>(
        frame,
        (const float*)d_in[1], (const float*)d_in[2], (const float*)d_in[3],
        (const float*)d_in[4], (const float*)d_in[5],
        (float*)d_out);
}